// WelfordEstimator_12043088298323
// MI455X (gfx1250) — compile-verified
//
#include <hip/hip_runtime.h>
#include <hip/hip_bf16.h>

// Welford online mean/M2 over the batch dim of x[B=64, C=256, H=56, W=56] f32.
// Output: out[0 .. F)  = running mean m
//         out[F .. 2F) = M2 accumulator s        (F = C*H*W = 802816)
//
// Roofline: 205.5 MB read (once) + 6.4 MB write @ 23.3 TB/s -> ~9.1 us floor;
// ~0.2 G lane-ops of VALU -> irrelevant. Welford is a sequential scan with no
// matmul structure, so WMMA does not apply; the MI455X-relevant path is pure
// data movement: coalesced non-temporal global_load_b128 + global_prefetch_b8.
//
// v2 changes vs v1:
//  - batch stride is constexpr (shape fixed by harness) -> per-iteration
//    addresses fold to constant adds instead of v_mad_nc_i64_i32 chains.
//  - prefetch locality 3 (nearest scope) and distance 8 slices for a longer
//    latency lead; always in-bounds so non-speculative prefetch is safe.
//  - grid divides NVEC exactly (200704 = 784 * 256) -> no bounds branch.

typedef float v4f __attribute__((ext_vector_type(4)));

constexpr int WELF_B = 64;
constexpr int FEAT   = 256 * 56 * 56;  // 802816 floats per batch slice
constexpr int NVEC   = FEAT / 4;       // 200704 float4s per batch slice
constexpr int BLOCK  = 256;            // 8 wave32s per workgroup
constexpr int GRID   = NVEC / BLOCK;   // 784, exact
constexpr int PF     = 8;              // prefetch distance in batch slices

__global__ __launch_bounds__(BLOCK) void welford_b64_kernel(
    const float* __restrict__ x, float* __restrict__ out) {
  const int i = blockIdx.x * BLOCK + threadIdx.x;  // always < NVEC

  const v4f* __restrict__ xv = reinterpret_cast<const v4f*>(x) + i;

  v4f m = (v4f)(0.0f);
  v4f s = (v4f)(0.0f);

#pragma unroll
  for (int b = 0; b < WELF_B; ++b) {
    if (b + PF < WELF_B) {
      // gfx1250 global_prefetch_b8: warm the strided slice PF iterations
      // ahead while the dependent Welford chain executes. Guarded in-bounds.
      __builtin_prefetch((const void*)(xv + (size_t)(b + PF) * NVEC), 0, 3);
    }
    // Each element read exactly once -> non-temporal b128 load.
    v4f xi = __builtin_nontemporal_load(xv + (size_t)b * NVEC);

    // Fully unrolled: rn is a correctly-rounded compile-time constant.
    const float rn = 1.0f / (float)(b + 1);

    float d;
    d = xi.x - m.x; m.x += d * rn; s.x += d * (xi.x - m.x);
    d = xi.y - m.y; m.y += d * rn; s.y += d * (xi.y - m.y);
    d = xi.z - m.z; m.z += d * rn; s.z += d * (xi.z - m.z);
    d = xi.w - m.w; m.w += d * rn; s.w += d * (xi.w - m.w);
  }

  v4f* __restrict__ om = reinterpret_cast<v4f*>(out) + i;
  v4f* __restrict__ os = reinterpret_cast<v4f*>(out) + NVEC + i;
  __builtin_nontemporal_store(m, om);
  __builtin_nontemporal_store(s, os);
}

extern "C" void kernel_launch(void* const* d_in, const int* in_sizes, int n_in,
                              void* d_out, int out_size, void* d_ws, size_t ws_size,
                              hipStream_t stream) {
  const float* x = (const float*)d_in[0];
  float* out = (float*)d_out;

  welford_b64_kernel<<<GRID, BLOCK, 0, stream>>>(x, out);
}